// SlidingWindowPCC_2989297238153
// MI455X (gfx1250) — compile-verified
//
#include <hip/hip_runtime.h>

typedef __attribute__((ext_vector_type(2))) float v2f;
typedef __attribute__((ext_vector_type(4))) float v4f;
typedef __attribute__((ext_vector_type(8))) float v8f;

#define WINDOW   128
#define WSTRIDE  16
#define T_DIM    4096
#define F_DIM    128
#define NWIN     249   // (4096-128)/16 + 1

// Bank-conflict-avoiding LDS swizzle:
// A/B fragment loads have lanes 0-15 on row r and lanes 16-31 on row r+2
// (rows differ in bit 1). Shifting columns by 32 floats when (row & 2)
// puts the two 16-wide lane groups in disjoint bank halves. Keeps LDS at
// exactly 128*128 floats = 64 KB.
__device__ __forceinline__ int lds_idx(int row, int col) {
    return row * F_DIM + ((col + ((row & 2) << 4)) & (F_DIM - 1));
}

__global__ __launch_bounds__(256)
void SlidingWindowPCC_kernel(const float* __restrict__ in,
                             float* __restrict__ out) {
    __shared__ float lds[F_DIM * WINDOW];   // 64 KB

    const int w   = blockIdx.x;             // window index
    const int b   = blockIdx.y;             // batch index
    const int tid = threadIdx.x;

    // ---- Stage the 128x128 fp32 window into LDS (coalesced b128 loads) ----
    const float* src = in + ((size_t)b * T_DIM + (size_t)w * WSTRIDE) * F_DIM;
    #pragma unroll
    for (int i = 0; i < 16; ++i) {
        int idx = tid + i * 256;            // 0..4095 float4 slots
        int row = idx >> 5;                 // 32 float4 per row
        int col = (idx & 31) << 2;
        v4f val = *(const v4f*)(src + row * F_DIM + col);
        *(v4f*)(lds + lds_idx(row, col)) = val;   // swizzle shifts whole rows; stays 16B-aligned
    }
    __syncthreads();

    // ---- Per-wave 2x4 block of 16x16 tiles over the 8x8 tile grid ----
    const int wave = tid >> 5;
    const int lane = tid & 31;
    const int half = lane >> 4;             // 0: lanes 0-15, 1: lanes 16-31
    const int ln   = lane & 15;
    const int fb   = ((wave & 3) << 1) << 4;   // f tile base: waves 0..3 -> rows 0,32,64,96 (2 tiles each)
    const int gb   = ((wave >> 2) << 2) << 4;  // g tile base: waves /4    -> cols 0,64      (4 tiles each)

    v8f acc[2][4];
    #pragma unroll
    for (int i = 0; i < 2; ++i)
        #pragma unroll
        for (int j = 0; j < 4; ++j)
            acc[i][j] = (v8f){0.f, 0.f, 0.f, 0.f, 0.f, 0.f, 0.f, 0.f};

    // A(16x4) fragment: element [m = ln][k = 2*half + v]  (v = component 0/1)
    // B(4x16) fragment: element [k = 2*half + v][n = ln]
    // Gram: A[m][k] = X[l][f] (transpose), B[k][n] = X[l][g] -> same LDS row,
    // different column bases.
    #pragma unroll 4
    for (int k0 = 0; k0 < WINDOW; k0 += 4) {
        const int r0 = k0 + 2 * half;       // rows r0, r0+1 for components .x/.y

        v2f a0, a1, bb0, bb1, bb2, bb3;
        a0.x  = lds[lds_idx(r0,     fb      + ln)];
        a0.y  = lds[lds_idx(r0 + 1, fb      + ln)];
        a1.x  = lds[lds_idx(r0,     fb + 16 + ln)];
        a1.y  = lds[lds_idx(r0 + 1, fb + 16 + ln)];
        bb0.x = lds[lds_idx(r0,     gb      + ln)];
        bb0.y = lds[lds_idx(r0 + 1, gb      + ln)];
        bb1.x = lds[lds_idx(r0,     gb + 16 + ln)];
        bb1.y = lds[lds_idx(r0 + 1, gb + 16 + ln)];
        bb2.x = lds[lds_idx(r0,     gb + 32 + ln)];
        bb2.y = lds[lds_idx(r0 + 1, gb + 32 + ln)];
        bb3.x = lds[lds_idx(r0,     gb + 48 + ln)];
        bb3.y = lds[lds_idx(r0 + 1, gb + 48 + ln)];

        // 8 args: (neg_a, A, neg_b, B, c_mod, C, reuse_a, reuse_b)
        acc[0][0] = __builtin_amdgcn_wmma_f32_16x16x4_f32(false, a0, false, bb0, (short)0, acc[0][0], false, false);
        acc[0][1] = __builtin_amdgcn_wmma_f32_16x16x4_f32(false, a0, false, bb1, (short)0, acc[0][1], false, false);
        acc[0][2] = __builtin_amdgcn_wmma_f32_16x16x4_f32(false, a0, false, bb2, (short)0, acc[0][2], false, false);
        acc[0][3] = __builtin_amdgcn_wmma_f32_16x16x4_f32(false, a0, false, bb3, (short)0, acc[0][3], false, false);
        acc[1][0] = __builtin_amdgcn_wmma_f32_16x16x4_f32(false, a1, false, bb0, (short)0, acc[1][0], false, false);
        acc[1][1] = __builtin_amdgcn_wmma_f32_16x16x4_f32(false, a1, false, bb1, (short)0, acc[1][1], false, false);
        acc[1][2] = __builtin_amdgcn_wmma_f32_16x16x4_f32(false, a1, false, bb2, (short)0, acc[1][2], false, false);
        acc[1][3] = __builtin_amdgcn_wmma_f32_16x16x4_f32(false, a1, false, bb3, (short)0, acc[1][3], false, false);
    }

    // ---- Store: D VGPR v holds [m = v + 8*half][n = ln]; scale by 1/WINDOW ----
    float* dst = out + (size_t)(b * NWIN + w) * (F_DIM * F_DIM);
    const float scale = 1.0f / (float)WINDOW;   // 0.0078125
    #pragma unroll
    for (int i = 0; i < 2; ++i) {
        #pragma unroll
        for (int j = 0; j < 4; ++j) {
            #pragma unroll
            for (int v = 0; v < 8; ++v) {
                int m = fb + i * 16 + v + 8 * half;
                int n = gb + j * 16 + ln;
                dst[m * F_DIM + n] = acc[i][j][v] * scale;
            }
        }
    }
}

extern "C" void kernel_launch(void* const* d_in, const int* in_sizes, int n_in,
                              void* d_out, int out_size, void* d_ws, size_t ws_size,
                              hipStream_t stream) {
    const float* in = (const float*)d_in[0];
    float* out = (float*)d_out;
    dim3 grid(NWIN, 16);     // 249 windows x 16 batches = 3984 workgroups
    dim3 block(256);         // 8 wave32
    SlidingWindowPCC_kernel<<<grid, block, 0, stream>>>(in, out);
}